// RNN_6717328851773
// MI455X (gfx1250) — compile-verified
//
#include <hip/hip_runtime.h>
#include <math.h>

typedef __attribute__((ext_vector_type(2))) float v2f;
typedef __attribute__((ext_vector_type(8))) float v8f;

#define TT 512
#define BB 1024
#define HH 31      // hidden size
#define G3 93      // 3*H
#define GP 96      // 3*H padded to 6 N-tiles of 16
#define KH 32      // hidden K padded to multiple of 4
#define NTILES 6   // GP/16

__device__ __forceinline__ float sigmoidf_(float v) {
    return 1.0f / (1.0f + __expf(-v));
}

// One wave (32 lanes) per block; each block owns a 16-row batch tile and runs
// the full T-step recurrence. DINA/DINB are the *padded* per-source feature
// counts (layer0: 36/0 ; layer1: 32/32 reading y0f,y0b which are zero-padded
// at column 31).
template<int DINA, int DINB>
__global__ __launch_bounds__(32)
void gru_dir_kernel(const float* __restrict__ xA, const float* __restrict__ xB,
                    const float* __restrict__ Wih, const float* __restrict__ Whh,
                    const float* __restrict__ bih, const float* __restrict__ bhh,
                    float* __restrict__ y,        // (B,T,KH) padded, or null
                    float* __restrict__ hlast,    // (B,KH) captured at orig t=T-1, or null
                    int reverse) {
    constexpr int DIN  = DINA + DINB;       // padded total input features
    constexpr int KA   = DINA / 4;
    constexpr int KB   = DINB / 4;
    constexpr int DINW = (DINB > 0) ? 62 : 36;  // actual W_ih column count

    __shared__ float wih_lds[DIN * GP];   // [k][g], zero-padded
    __shared__ float whh_lds[KH * GP];    // [k][g], zero-padded
    __shared__ float h_lds[16 * KH];      // row-major h tile == A-layout source
    __shared__ float gx_lds[16 * GP];     // x-gates (incl. b_ih)
    __shared__ float gh_lds[16 * GP];     // h-gates (incl. b_hh)

    const int lane = threadIdx.x;
    const int lo   = lane & 15;
    const int hi   = lane >> 4;
    const int b0   = blockIdx.x * 16;

    // --- load W_ih into LDS in B-matrix ([k][g]) form, zero padding ---
    for (int i = lane; i < DIN * GP; i += 32) {
        const int k = i / GP, g = i % GP;
        float w = 0.0f;
        int  ka = k;
        bool ok = true;
        if (DINB > 0) {                    // concat(y0f[0:31], y0b[0:31]) padded to 64
            if (k == 31 || k == 63) ok = false;
            else ka = (k < 31) ? k : (k - 1);
        }
        if (ok && g < G3) w = Wih[g * DINW + ka];
        wih_lds[i] = w;
    }
    // --- W_hh ---
    for (int i = lane; i < KH * GP; i += 32) {
        const int k = i / GP, g = i % GP;
        whh_lds[i] = (k < HH && g < G3) ? Whh[g * HH + k] : 0.0f;
    }
    // --- h0 = 0 ---
    for (int i = lane; i < 16 * KH; i += 32) h_lds[i] = 0.0f;

    // --- per-lane broadcast biases (depend only on gate column) ---
    float bihv[NTILES], bhhv[NTILES];
#pragma unroll
    for (int nt = 0; nt < NTILES; ++nt) {
        const int g = nt * 16 + lo;
        bihv[nt] = (g < G3) ? bih[g] : 0.0f;
        bhhv[nt] = (g < G3) ? bhh[g] : 0.0f;
    }
    __syncthreads();

    for (int s = 0; s < TT; ++s) {
        const int t = reverse ? (TT - 1 - s) : s;

        v8f cx[NTILES], ch[NTILES];
#pragma unroll
        for (int nt = 0; nt < NTILES; ++nt)
#pragma unroll
            for (int v = 0; v < 8; ++v) { cx[nt][v] = bihv[nt]; ch[nt][v] = bhhv[nt]; }

        // ---- xg = x_t @ Wih^T + b_ih   (off the serial critical path) ----
#pragma unroll
        for (int kt = 0; kt < KA + KB; ++kt) {
            v2f a;
            if constexpr (DINB > 0) {
                if (kt < KA) {
                    const float* p = xA + ((size_t)(b0 + lo) * TT + t) * DINA + kt * 4 + 2 * hi;
                    a.x = p[0]; a.y = p[1];
                } else {
                    const float* p = xB + ((size_t)(b0 + lo) * TT + t) * DINB + (kt - KA) * 4 + 2 * hi;
                    a.x = p[0]; a.y = p[1];
                }
            } else {
                const float* p = xA + ((size_t)(b0 + lo) * TT + t) * DINA + kt * 4 + 2 * hi;
                a.x = p[0]; a.y = p[1];
            }
#pragma unroll
            for (int nt = 0; nt < NTILES; ++nt) {
                const float* wp = &wih_lds[(kt * 4 + 2 * hi) * GP + nt * 16 + lo];
                v2f b; b.x = wp[0]; b.y = wp[GP];
                cx[nt] = __builtin_amdgcn_wmma_f32_16x16x4_f32(
                    false, a, false, b, (short)0, cx[nt], false, false);
            }
        }

        // ---- hg = h @ Whh^T + b_hh   (serial dependency on h_lds) ----
#pragma unroll
        for (int kt = 0; kt < KH / 4; ++kt) {
            const int kk = kt * 4 + 2 * hi;
            v2f a; a.x = h_lds[lo * KH + kk]; a.y = h_lds[lo * KH + kk + 1];
#pragma unroll
            for (int nt = 0; nt < NTILES; ++nt) {
                const float* wp = &whh_lds[kk * GP + nt * 16 + lo];
                v2f b; b.x = wp[0]; b.y = wp[GP];
                ch[nt] = __builtin_amdgcn_wmma_f32_16x16x4_f32(
                    false, a, false, b, (short)0, ch[nt], false, false);
            }
        }

        // ---- D-layout -> LDS (transpose for elementwise) ----
#pragma unroll
        for (int nt = 0; nt < NTILES; ++nt)
#pragma unroll
            for (int v = 0; v < 8; ++v) {
                const int row = v + 8 * hi;
                gx_lds[row * GP + nt * 16 + lo] = cx[nt][v];
                gh_lds[row * GP + nt * 16 + lo] = ch[nt][v];
            }
        __syncthreads();

        // ---- GRU cell: lane j handles hidden unit j for all 16 rows ----
        const int j = lane;
        for (int row = 0; row < 16; ++row) {
            float hnew = 0.0f;
            if (j < HH) {
                const float xr = gx_lds[row * GP + j];
                const float hr = gh_lds[row * GP + j];
                const float xz = gx_lds[row * GP + HH + j];
                const float hz = gh_lds[row * GP + HH + j];
                const float xn = gx_lds[row * GP + 2 * HH + j];
                const float hn = gh_lds[row * GP + 2 * HH + j];
                const float r = sigmoidf_(xr + hr);
                const float z = sigmoidf_(xz + hz);
                const float n = tanhf(xn + r * hn);
                const float hp = h_lds[row * KH + j];
                hnew = (1.0f - z) * n + z * hp;
            }
            h_lds[row * KH + j] = hnew;   // also zeroes pad column 31
            if (y)
                y[((size_t)(b0 + row) * TT + t) * KH + j] = hnew;
            if (hlast && t == TT - 1)
                hlast[(size_t)(b0 + row) * KH + j] = hnew;
        }
        __syncthreads();

        // layer-1 kernels (y==null) only need h at orig t==T-1:
        // forward: that's the final step anyway; backward: scan step 0 -> stop.
        if (!y && t == TT - 1) break;
    }
}

// out[b][c] = fc_b[c] + sum_j h1f[b][j]*fc_w[c][j] + h1b[b][j]*fc_w[c][31+j]
__global__ void fc_kernel(const float* __restrict__ hf, const float* __restrict__ hb,
                          const float* __restrict__ fcw, const float* __restrict__ fcb,
                          float* __restrict__ out) {
    const int b = blockIdx.x * blockDim.x + threadIdx.x;
    if (b >= BB) return;
    float acc[7];
#pragma unroll
    for (int c = 0; c < 7; ++c) acc[c] = fcb[c];
    for (int jj = 0; jj < HH; ++jj) {
        const float vf = hf[(size_t)b * KH + jj];
        const float vb = hb[(size_t)b * KH + jj];
#pragma unroll
        for (int c = 0; c < 7; ++c)
            acc[c] += vf * fcw[c * 62 + jj] + vb * fcw[c * 62 + HH + jj];
    }
#pragma unroll
    for (int c = 0; c < 7; ++c) out[(size_t)b * 7 + c] = acc[c];
}

extern "C" void kernel_launch(void* const* d_in, const int* in_sizes, int n_in,
                              void* d_out, int out_size, void* d_ws, size_t ws_size,
                              hipStream_t stream) {
    (void)in_sizes; (void)n_in; (void)out_size; (void)ws_size;
    const float* x       = (const float*)d_in[0];
    const float* Wih_l0f = (const float*)d_in[1];
    const float* Whh_l0f = (const float*)d_in[2];
    const float* bih_l0f = (const float*)d_in[3];
    const float* bhh_l0f = (const float*)d_in[4];
    const float* Wih_l0b = (const float*)d_in[5];
    const float* Whh_l0b = (const float*)d_in[6];
    const float* bih_l0b = (const float*)d_in[7];
    const float* bhh_l0b = (const float*)d_in[8];
    const float* Wih_l1f = (const float*)d_in[9];
    const float* Whh_l1f = (const float*)d_in[10];
    const float* bih_l1f = (const float*)d_in[11];
    const float* bhh_l1f = (const float*)d_in[12];
    const float* Wih_l1b = (const float*)d_in[13];
    const float* Whh_l1b = (const float*)d_in[14];
    const float* bih_l1b = (const float*)d_in[15];
    const float* bhh_l1b = (const float*)d_in[16];
    const float* fc_w    = (const float*)d_in[17];
    const float* fc_b    = (const float*)d_in[18];

    float* ws  = (float*)d_ws;
    float* y0f = ws;                                 // B*T*KH
    float* y0b = y0f + (size_t)BB * TT * KH;         // B*T*KH
    float* h1f = y0b + (size_t)BB * TT * KH;         // B*KH
    float* h1b = h1f + (size_t)BB * KH;              // B*KH

    const dim3 grid(BB / 16), block(32);
    // layer 0 (stores full sequence, zero-padded at col 31)
    gru_dir_kernel<36, 0><<<grid, block, 0, stream>>>(
        x, nullptr, Wih_l0f, Whh_l0f, bih_l0f, bhh_l0f, y0f, nullptr, 0);
    gru_dir_kernel<36, 0><<<grid, block, 0, stream>>>(
        x, nullptr, Wih_l0b, Whh_l0b, bih_l0b, bhh_l0b, y0b, nullptr, 1);
    // layer 1 (only h at original t = T-1 is needed)
    gru_dir_kernel<32, 32><<<grid, block, 0, stream>>>(
        y0f, y0b, Wih_l1f, Whh_l1f, bih_l1f, bhh_l1f, nullptr, h1f, 0);
    gru_dir_kernel<32, 32><<<grid, block, 0, stream>>>(
        y0f, y0b, Wih_l1b, Whh_l1b, bih_l1b, bhh_l1b, nullptr, h1b, 1);
    // final classifier
    fc_kernel<<<(BB + 255) / 256, 256, 0, stream>>>(h1f, h1b, fc_w, fc_b, (float*)d_out);
}